// LayerRGCN_31155692765354
// MI455X (gfx1250) — compile-verified
//
#include <hip/hip_runtime.h>

typedef float v2f __attribute__((ext_vector_type(2)));
typedef float v8f __attribute__((ext_vector_type(8)));

#define NNODES 50000
#define NEDGES 800000
#define RELN   8
#define FIN    64
#define FOUT   64
#define ROWT   (NNODES / 16)          // 3125 row tiles (exact)
#define COLT   (FOUT / 16)            // 4 col tiles
#define APAD   68                     // LDS row stride (floats): bank = (4m+k)%64

// Async stage of a 16x64 f32 tile (4 KB) into LDS with padded rows.
// Each calling thread copies one 16B segment: seg s of row r.
// LDS destination operand must be a single 32-bit VGPR (LDS byte address):
// truncate the generic shared pointer to its low DWORD.
__device__ __forceinline__ void stage_tile_async(
    float* sA, const float* gsrc, int tid)
{
    const int row = tid >> 4;          // 0..15
    const int seg = tid & 15;          // 0..15 (16B segments of a 64-float row)
    const float* g = gsrc + row * FIN + seg * 4;
    const unsigned lds_addr = (unsigned)(uintptr_t)(sA + row * APAD + seg * 4);
    asm volatile("global_load_async_to_lds_b128 %0, %1, off"
                 :: "v"(lds_addr), "v"(g) : "memory");
}

__device__ __forceinline__ void wait_async_all()
{
    asm volatile("s_wait_asynccnt 0x0" ::: "memory");
}

// ---------------------------------------------------------------------------
// K1: transformed[n, r, o] = sum_f h[n,f] * weight[r,f,o]
// One block (8 waves) per row tile. A tile staged once via async-to-LDS;
// wave w = relation w, loops 4 col tiles with A fragments register-resident.
// 512 WMMAs per block.
// ---------------------------------------------------------------------------
__global__ __launch_bounds__(256) void rgcn_transform_wmma(
    const float* __restrict__ h,
    const float* __restrict__ weight,
    float* __restrict__ transformed)
{
    __shared__ float sA[16 * APAD];

    const int rt   = blockIdx.x;                 // row tile 0..3124
    const int lane = threadIdx.x & 31;
    const int r    = threadIdx.x >> 5;           // wave id == relation

    stage_tile_async(sA, h + (size_t)rt * 16 * FIN, threadIdx.x);
    wait_async_all();
    __syncthreads();

    const int m  = lane & 15;                    // row-in-tile (A) / col (B)
    const int hi = lane >> 4;                    // lane-half selector

    // A fragments from LDS (ds_load_b64, conflict-free via APAD padding),
    // loaded once and reused across all 4 col tiles.
    v2f afrag[FIN / 4];
#pragma unroll
    for (int kb = 0; kb < FIN / 4; ++kb)
        afrag[kb] = *(const v2f*)&sA[m * APAD + kb * 4 + hi * 2];

    const float* wrel = weight + r * FIN * FOUT;

#pragma unroll
    for (int ct = 0; ct < COLT; ++ct) {
        const float* wbase = wrel + ct * 16 + m;
        v8f c = {};
#pragma unroll
        for (int kb = 0; kb < FIN / 4; ++kb) {
            const int k0 = kb * 4;
            v2f b;                                // B 4x16: VGPR v = row K=v / K=v+2
            b.x = wbase[(k0 + hi * 2    ) * FOUT];
            b.y = wbase[(k0 + hi * 2 + 1) * FOUT];
            c = __builtin_amdgcn_wmma_f32_16x16x4_f32(
                    false, afrag[kb], false, b, (short)0, c, false, false);
        }
        // D layout: VGPR v -> M = v + 8*hi, N = lane&15
        const int col = ct * 16 + m;
#pragma unroll
        for (int v = 0; v < 8; ++v) {
            const int row = rt * 16 + v + hi * 8;
            transformed[row * (RELN * FOUT) + r * FOUT + col] = c[v];
        }
    }
}

// ---------------------------------------------------------------------------
// K2: zero the h_sum accumulator
// ---------------------------------------------------------------------------
__global__ __launch_bounds__(256) void rgcn_zero(float* __restrict__ p, int n)
{
    const int i = blockIdx.x * 256 + threadIdx.x;
    if (i < n) p[i] = 0.0f;
}

// ---------------------------------------------------------------------------
// K3: per-edge gather + scale + scatter-add
// One wave per edge: 32 lanes x float2 = 64 features, contiguous 256B gather
// (L2-resident), two non-returning global_atomic_add_f32 per lane.
// ---------------------------------------------------------------------------
__global__ __launch_bounds__(256) void rgcn_edge_scatter(
    const float* __restrict__ transformed,
    const int* __restrict__ edge_src,
    const int* __restrict__ edge_dst,
    const int* __restrict__ rel_type,
    const float* __restrict__ norm,
    float* __restrict__ hsum)
{
    const int t    = blockIdx.x * 256 + threadIdx.x;
    const int e    = t >> 5;                     // 800000 edges (exact grid)
    const int lane = t & 31;

    const int   src = edge_src[e];
    const int   dst = edge_dst[e];
    const int   rel = rel_type[e];
    const float nrm = norm[e];

    const int o = lane * 2;
    v2f v = *(const v2f*)(transformed + src * (RELN * FOUT) + rel * FOUT + o);

    float* d = hsum + dst * FOUT + o;
    unsafeAtomicAdd(d,     v.x * nrm);
    unsafeAtomicAdd(d + 1, v.y * nrm);
}

// ---------------------------------------------------------------------------
// K4: out = relu(h @ W_0 + h_sum)
// One block (4 waves) per row tile; A via async-to-LDS; wave w = col tile w.
// ---------------------------------------------------------------------------
__global__ __launch_bounds__(128) void rgcn_out_wmma(
    const float* __restrict__ h,
    const float* __restrict__ W0,
    const float* __restrict__ hsum,
    float* __restrict__ out)
{
    __shared__ float sA[16 * APAD];

    const int rt   = blockIdx.x;                 // 0..3124
    const int lane = threadIdx.x & 31;
    const int ct   = threadIdx.x >> 5;           // wave id == col tile

    // 128 threads copy 4KB in two 16B segments each.
    stage_tile_async(sA, h + (size_t)rt * 16 * FIN, threadIdx.x);
    stage_tile_async(sA, h + (size_t)rt * 16 * FIN, threadIdx.x + 128);
    wait_async_all();
    __syncthreads();

    const int m  = lane & 15;
    const int hi = lane >> 4;

    const float* wbase = W0 + ct * 16 + m;

    v8f c = {};
#pragma unroll
    for (int kb = 0; kb < FIN / 4; ++kb) {
        const int k0 = kb * 4;
        v2f a = *(const v2f*)&sA[m * APAD + k0 + hi * 2];
        v2f b;
        b.x = wbase[(k0 + hi * 2    ) * FOUT];
        b.y = wbase[(k0 + hi * 2 + 1) * FOUT];
        c = __builtin_amdgcn_wmma_f32_16x16x4_f32(
                false, a, false, b, (short)0, c, false, false);
    }

    const int col = ct * 16 + m;
#pragma unroll
    for (int v = 0; v < 8; ++v) {
        const int row = rt * 16 + v + hi * 8;
        const float x = c[v] + hsum[row * FOUT + col];
        out[row * FOUT + col] = fmaxf(x, 0.0f);
    }
}

// ---------------------------------------------------------------------------
extern "C" void kernel_launch(void* const* d_in, const int* in_sizes, int n_in,
                              void* d_out, int out_size, void* d_ws, size_t ws_size,
                              hipStream_t stream)
{
    const float* h        = (const float*)d_in[0];
    const int*   edge_src = (const int*)  d_in[1];
    const int*   edge_dst = (const int*)  d_in[2];
    const int*   rel_type = (const int*)  d_in[3];
    const float* norm     = (const float*)d_in[4];
    const float* weight   = (const float*)d_in[5];
    const float* W0       = (const float*)d_in[6];
    float*       out      = (float*)d_out;

    // Workspace: transformed [50000*8*64] f32 (102.4 MB) then hsum [50000*64].
    float* transformed = (float*)d_ws;
    float* hsum        = transformed + (size_t)NNODES * RELN * FOUT;

    // K1: one block per row tile (3125 blocks, 8 waves = 8 relations each).
    rgcn_transform_wmma<<<ROWT, 256, 0, stream>>>(h, weight, transformed);

    // K2: zero hsum (50000*64 = 3.2M floats -> 12500 blocks).
    const int nsum = NNODES * FOUT;
    rgcn_zero<<<(nsum + 255) / 256, 256, 0, stream>>>(hsum, nsum);

    // K3: 800000 edges * 32 lanes = 25.6M threads -> 100000 blocks.
    rgcn_edge_scatter<<<(NEDGES * 32) / 256, 256, 0, stream>>>(
        transformed, edge_src, edge_dst, rel_type, norm, hsum);

    // K4: one block per row tile (3125 blocks, 4 waves = 4 col tiles each).
    rgcn_out_wmma<<<ROWT, 128, 0, stream>>>(h, W0, hsum, out);
}